// MTAD_GAT_31911607009385
// MI455X (gfx1250) — compile-verified
//
#include <hip/hip_runtime.h>
#include <hip/hip_bf16.h>

// ---------------- model dims ----------------
#define MB   64     // batch
#define MW   128    // window
#define MK   64     // features
#define MKS  7      // conv kernel
#define MEF  128    // = W
#define MET  128    // = 2*K
#define MH   256    // GRU hidden
#define MFH  256    // fc hidden

typedef float v2f __attribute__((ext_vector_type(2)));
typedef float v8f __attribute__((ext_vector_type(8)));

__device__ __forceinline__ float sigf(float x) { return 1.f / (1.f + __expf(-x)); }

__device__ __forceinline__ void wait_asynccnt0()
{
#if __has_builtin(__builtin_amdgcn_s_wait_asynccnt)
    __builtin_amdgcn_s_wait_asynccnt(0);
#else
    asm volatile("s_wait_asynccnt 0x0" ::: "memory");
#endif
}

// =====================================================================
// Generic fp32 WMMA GEMM:  C[m,n] = act( sum_k A[m*lda+k]*Bw[n*ldb+k] + bias[n] )
// Register-blocked: each wave computes a 32(M) x 64(N) tile = 8 accumulators,
// 8 x v_wmma_f32_16x16x4_f32 per k-step with 2 A-frags / 4 B-frags (A,B reuse).
// grid.x = N/64, grid.y = M/32, block = 32 (one wave, EXEC all ones).
// M multiple of 32, N multiple of 64, Kd multiple of 4.
// =====================================================================
__global__ void gemm_wmma_f32(const float* __restrict__ A, int lda,
                              const float* __restrict__ Bw, int ldb,
                              const float* __restrict__ bias,
                              float* __restrict__ C, int ldc,
                              int Kd, int act)
{
    const int lane = threadIdx.x & 31;
    const int idx  = lane & 15;          // M (for A) / N (for B/C)
    const int half = lane >> 4;          // 0 or 1
    const int kh   = half * 2;           // K sub-offset within x4 step
    const int row0 = blockIdx.y * 32;
    const int col0 = blockIdx.x * 64;

    const float* __restrict__ Ar[2];
    Ar[0] = A + (row0 + idx) * lda;
    Ar[1] = Ar[0] + 16 * lda;
    const float* __restrict__ Br[4];
    Br[0] = Bw + (col0 + idx) * ldb;
    Br[1] = Br[0] + 16 * ldb;
    Br[2] = Br[0] + 32 * ldb;
    Br[3] = Br[0] + 48 * ldb;

    v8f acc[2][4] = {};
    for (int k0 = 0; k0 < Kd; k0 += 4) {
        // speculative prefetch of upcoming K-lines (global_prefetch_b8);
        // OOB prefetches are dropped by hardware.
        __builtin_prefetch(Ar[0] + k0 + 64, 0, 1);
        __builtin_prefetch(Br[0] + k0 + 64, 0, 1);

        v2f a[2], b[4];
#pragma unroll
        for (int i = 0; i < 2; ++i) {
            a[i].x = Ar[i][k0 + kh];  a[i].y = Ar[i][k0 + kh + 1];
        }
#pragma unroll
        for (int j = 0; j < 4; ++j) {
            b[j].x = Br[j][k0 + kh];  b[j].y = Br[j][k0 + kh + 1];
        }
#pragma unroll
        for (int i = 0; i < 2; ++i)
#pragma unroll
            for (int j = 0; j < 4; ++j)
                acc[i][j] = __builtin_amdgcn_wmma_f32_16x16x4_f32(
                                false, a[i], false, b[j], (short)0,
                                acc[i][j], false, false);
    }

#pragma unroll
    for (int i = 0; i < 2; ++i) {
#pragma unroll
        for (int j = 0; j < 4; ++j) {
            const int n  = col0 + j * 16 + idx;
            const float bv = bias ? bias[n] : 0.0f;
#pragma unroll
            for (int r = 0; r < 8; ++r) {
                int m   = row0 + i * 16 + half * 8 + r;
                float v = acc[i][j][r] + bv;
                if (act == 1) v = v > 0.f ? v : 0.f;
                C[m * ldc + n] = v;
            }
        }
    }
}

// =====================================================================
// Conv1d (pad=3, cross-correlation) + ReLU; writes y and packs z[:, :64].
// Weights staged into LDS via GLOBAL_LOAD_ASYNC_TO_LDS_B32 (ASYNCcnt path).
// grid = (K_out, B), block = W threads
// =====================================================================
__global__ void conv_relu_pack(const float* __restrict__ x,
                               const float* __restrict__ cw,
                               const float* __restrict__ cb,
                               float* __restrict__ y, float* __restrict__ z)
{
    __shared__ float wsh[MK * MKS];
    const int o = blockIdx.x, b = blockIdx.y, t = threadIdx.x;

    // async copy cw[o,:,:] (448 floats) into LDS
    const float* gsrc = cw + o * MK * MKS;
    for (int i = t; i < MK * MKS; i += MW) {
        unsigned lds_off = (unsigned)(unsigned long long)&wsh[i];
        unsigned long long ga = (unsigned long long)(gsrc + i);
        asm volatile("global_load_async_to_lds_b32 %0, %1, off"
                     :: "v"(lds_off), "v"(ga) : "memory");
    }
    wait_asynccnt0();      // each wave waits for its own async loads
    __syncthreads();       // publish LDS across the workgroup

    const float* xb = x + b * MW * MK;
    float acc = cb[o];
    for (int s = 0; s < MKS; ++s) {
        int tt = t + s - (MKS - 1) / 2;
        if (tt >= 0 && tt < MW) {
            const float* xr = xb + tt * MK;
#pragma unroll 8
            for (int i = 0; i < MK; ++i) acc += xr[i] * wsh[i * MKS + s];
        }
    }
    float v = acc > 0.f ? acc : 0.f;
    y[(b * MW + t) * MK + o]       = v;
    z[(b * MW + t) * (3 * MK) + o] = v;
}

// xf[b,k,w] = y[b,w,k]
__global__ void transpose_y(const float* __restrict__ y, float* __restrict__ xf)
{
    int idx = blockIdx.x * blockDim.x + threadIdx.x;          // B*K*W
    if (idx >= MB * MK * MW) return;
    int w = idx % MW, k = (idx / MW) % MK, b = idx / (MW * MK);
    xf[idx] = y[(b * MW + w) * MK + k];
}

// s1/s2 from Wx rows
__global__ void feat_scores(const float* __restrict__ Wx, const float* __restrict__ fa,
                            float* __restrict__ s1, float* __restrict__ s2)
{
    int idx = blockIdx.x * blockDim.x + threadIdx.x;          // B*K
    if (idx >= MB * MK) return;
    const float* r = Wx + idx * MEF;
    float a = 0.f, c = 0.f;
#pragma unroll 8
    for (int e = 0; e < MEF; ++e) { a += r[e] * fa[e]; c += r[e] * fa[MEF + e]; }
    s1[idx] = a; s2[idx] = c;
}

// att_f row: leaky(s1[i]+s2[j]) + bias[i,j]; softmax over j (64)
__global__ void feat_att(const float* __restrict__ s1, const float* __restrict__ s2,
                         const float* __restrict__ fbias, float* __restrict__ att)
{
    __shared__ float sh[MK];
    int bi = blockIdx.x;               // b*64 + i
    int b = bi >> 6, i = bi & 63, j = threadIdx.x;
    float v = s1[bi] + s2[b * MK + j];
    v = v >= 0.f ? v : 0.2f * v;
    v += fbias[i * MK + j];
    sh[j] = v; __syncthreads();
    for (int off = 32; off > 0; off >>= 1) { if (j < off) sh[j] = fmaxf(sh[j], sh[j + off]); __syncthreads(); }
    float mx = sh[0]; __syncthreads();
    float e = __expf(v - mx);
    sh[j] = e; __syncthreads();
    for (int off = 32; off > 0; off >>= 1) { if (j < off) sh[j] += sh[j + off]; __syncthreads(); }
    att[bi * MK + j] = e / sh[0];
}

// h_feat = sigmoid(att_f @ xf), transposed into z[:, 64:128]
__global__ void feat_head(const float* __restrict__ att, const float* __restrict__ xf,
                          float* __restrict__ z)
{
    int idx = blockIdx.x * blockDim.x + threadIdx.x;          // (b,i,w)
    if (idx >= MB * MK * MW) return;
    int w = idx % MW, i = (idx / MW) % MK, b = idx / (MW * MK);
    const float* ar = att + (b * MK + i) * MK;
    const float* xb = xf + b * MK * MW + w;
    float acc = 0.f;
#pragma unroll 8
    for (int j = 0; j < MK; ++j) acc += ar[j] * xb[j * MW];
    z[(b * MW + w) * (3 * MK) + MK + i] = sigf(acc);
}

// e_t[b,i,j] = sum_e leaky(p[b,i,e]+q[b,j,e]+tb[e]) * ta[e] + tbias[i,j]
__global__ void temp_scores(const float* __restrict__ p, const float* __restrict__ q,
                            const float* __restrict__ tb, const float* __restrict__ ta,
                            const float* __restrict__ tbias, float* __restrict__ et)
{
    __shared__ float psh[MET], tbsh[MET], tash[MET];
    int bi = blockIdx.x;               // b*128 + i
    int b = bi >> 7, i = bi & 127, j = threadIdx.x;
    psh[j] = p[bi * MET + j]; tbsh[j] = tb[j]; tash[j] = ta[j];
    __syncthreads();
    const float* qr = q + (b * MW + j) * MET;
    float acc = 0.f;
#pragma unroll 4
    for (int e = 0; e < MET; ++e) {
        float v = psh[e] + qr[e] + tbsh[e];
        v = v >= 0.f ? v : 0.2f * v;
        acc += v * tash[e];
    }
    et[bi * MW + j] = acc + tbias[i * MW + j];
}

// in-place softmax over rows of length 128
__global__ void softmax128(float* __restrict__ a)
{
    __shared__ float sh[MW];
    int row = blockIdx.x, j = threadIdx.x;
    float v = a[row * MW + j];
    sh[j] = v; __syncthreads();
    for (int off = 64; off > 0; off >>= 1) { if (j < off) sh[j] = fmaxf(sh[j], sh[j + off]); __syncthreads(); }
    float mx = sh[0]; __syncthreads();
    float e = __expf(v - mx);
    sh[j] = e; __syncthreads();
    for (int off = 64; off > 0; off >>= 1) { if (j < off) sh[j] += sh[j + off]; __syncthreads(); }
    a[row * MW + j] = e / sh[0];
}

// h_temp = sigmoid(att_t @ y) -> z[:, 128:192]
__global__ void temp_head(const float* __restrict__ att, const float* __restrict__ y,
                          float* __restrict__ z)
{
    int idx = blockIdx.x * blockDim.x + threadIdx.x;          // (b,t,k)
    if (idx >= MB * MW * MK) return;
    int k = idx % MK, t = (idx / MK) % MW, b = idx / (MK * MW);
    const float* ar = att + (b * MW + t) * MW;
    const float* yb = y + b * MW * MK + k;
    float acc = 0.f;
#pragma unroll 4
    for (int j = 0; j < MW; ++j) acc += ar[j] * yb[j * MK];
    z[(b * MW + t) * (3 * MK) + 2 * MK + k] = sigf(acc);
}

__global__ void zero_f32(float* __restrict__ a, int n)
{
    int i = blockIdx.x * blockDim.x + threadIdx.x;
    if (i < n) a[i] = 0.f;
}

// GRU gate update (in-place on h), one thread per (b, hidden j)
__global__ void gru_gate(const float* __restrict__ gi, const float* __restrict__ gh,
                         float* __restrict__ h, int t)
{
    int idx = blockIdx.x * blockDim.x + threadIdx.x;          // B*H
    int b = idx >> 8, j = idx & 255;
    const float* gir = gi + (b * MW + t) * (3 * MH);
    const float* ghr = gh + b * (3 * MH);
    float r  = sigf(gir[j]          + ghr[j]);
    float zg = sigf(gir[MH + j]     + ghr[MH + j]);
    float n  = tanhf(gir[2 * MH + j] + r * ghr[2 * MH + j]);
    float hp = h[idx];
    h[idx] = (1.f - zg) * n + zg * hp;
}

// =====================================================================
extern "C" void kernel_launch(void* const* d_in, const int* in_sizes, int n_in,
                              void* d_out, int out_size, void* d_ws, size_t ws_size,
                              hipStream_t stream)
{
    const float* x          = (const float*)d_in[0];
    const float* conv_w     = (const float*)d_in[1];
    const float* conv_b     = (const float*)d_in[2];
    const float* feat_lin_w = (const float*)d_in[3];
    const float* feat_lin_b = (const float*)d_in[4];
    const float* feat_a     = (const float*)d_in[5];
    const float* feat_bias  = (const float*)d_in[6];
    const float* temp_lin_w = (const float*)d_in[7];
    const float* temp_lin_b = (const float*)d_in[8];
    const float* temp_a     = (const float*)d_in[9];
    const float* temp_bias  = (const float*)d_in[10];
    const float* gru_wih    = (const float*)d_in[11];
    const float* gru_whh    = (const float*)d_in[12];
    const float* gru_bih    = (const float*)d_in[13];
    const float* gru_bhh    = (const float*)d_in[14];
    const float* fc0_w      = (const float*)d_in[15];
    const float* fc0_b      = (const float*)d_in[16];
    const float* fc1_w      = (const float*)d_in[17];
    const float* fc1_b      = (const float*)d_in[18];
    const float* fc2_w      = (const float*)d_in[19];
    const float* fc2_b      = (const float*)d_in[20];
    const float* fc3_w      = (const float*)d_in[21];
    const float* fc3_b      = (const float*)d_in[22];
    float* out = (float*)d_out;

    // ---- workspace partition (floats) ----
    float* ws   = (float*)d_ws;
    float* y    = ws;                         // B*W*K      = 524288
    float* xf   = y    + MB * MW * MK;        // B*K*W      = 524288
    float* Wx   = xf   + MB * MK * MW;        // B*K*EF     = 524288
    float* s1   = Wx   + MB * MK * MEF;       // B*K        = 4096
    float* s2   = s1   + MB * MK;             // 4096
    float* attf = s2   + MB * MK;             // B*K*K      = 262144
    float* z    = attf + MB * MK * MK;        // B*W*3K     = 1572864
    float* p    = z    + MB * MW * 3 * MK;    // B*W*ET     = 1048576
    float* q    = p    + MB * MW * MET;       // 1048576
    float* attt = q    + MB * MW * MET;       // B*W*W      = 1048576
    float* gi   = attt + MB * MW * MW;        // B*W*3H     = 6291456
    float* gh   = gi   + MB * MW * 3 * MH;    // B*3H       = 49152
    float* h    = gh   + MB * 3 * MH;         // B*H        = 16384
    float* v0   = h    + MB * MH;             // B*FH       = 16384
    float* v1   = v0   + MB * MFH;            // 16384

    // 1) conv + relu, pack z[:, 0:64]  (async-LDS weight staging)
    conv_relu_pack<<<dim3(MK, MB), MW, 0, stream>>>(x, conv_w, conv_b, y, z);
    transpose_y<<<(MB * MK * MW + 255) / 256, 256, 0, stream>>>(y, xf);

    // 2) feature attention
    //    Wx = xf(4096x128) @ feat_lin_w(128x128)^T + b            [WMMA]
    gemm_wmma_f32<<<dim3(MEF / 64, (MB * MK) / 32), 32, 0, stream>>>(
        xf, MW, feat_lin_w, MW, feat_lin_b, Wx, MEF, MW, 0);
    feat_scores<<<(MB * MK + 255) / 256, 256, 0, stream>>>(Wx, feat_a, s1, s2);
    feat_att<<<MB * MK, MK, 0, stream>>>(s1, s2, feat_bias, attf);
    feat_head<<<(MB * MK * MW + 255) / 256, 256, 0, stream>>>(attf, xf, z);

    // 3) temporal attention
    //    p = y(8192x64) @ W1(128x64)^T ; q with W2                [WMMA]
    gemm_wmma_f32<<<dim3(MET / 64, (MB * MW) / 32), 32, 0, stream>>>(
        y, MK, temp_lin_w, 2 * MK, nullptr, p, MET, MK, 0);
    gemm_wmma_f32<<<dim3(MET / 64, (MB * MW) / 32), 32, 0, stream>>>(
        y, MK, temp_lin_w + MK, 2 * MK, nullptr, q, MET, MK, 0);
    temp_scores<<<MB * MW, MW, 0, stream>>>(p, q, temp_lin_b, temp_a, temp_bias, attt);
    softmax128<<<MB * MW, MW, 0, stream>>>(attt);
    temp_head<<<(MB * MW * MK + 255) / 256, 256, 0, stream>>>(attt, y, z);

    // 4) GRU input projection hoisted out of the scan:
    //    gi = z(8192x192) @ gru_wih(768x192)^T + bih              [WMMA, big]
    gemm_wmma_f32<<<dim3((3 * MH) / 64, (MB * MW) / 32), 32, 0, stream>>>(
        z, 3 * MK, gru_wih, 3 * MK, gru_bih, gi, 3 * MH, 3 * MK, 0);

    // 5) GRU scan: only the recurrent GEMM + gates per step
    zero_f32<<<(MB * MH + 255) / 256, 256, 0, stream>>>(h, MB * MH);
    for (int t = 0; t < MW; ++t) {
        gemm_wmma_f32<<<dim3((3 * MH) / 64, MB / 32), 32, 0, stream>>>(
            h, MH, gru_whh, MH, gru_bhh, gh, 3 * MH, MH, 0);
        gru_gate<<<(MB * MH) / 256, 256, 0, stream>>>(gi, gh, h, t);
    }

    // 6) MLP head                                                  [WMMA]
    gemm_wmma_f32<<<dim3(MFH / 64, MB / 32), 32, 0, stream>>>(
        h, MH, fc0_w, MH, fc0_b, v0, MFH, MH, 1);
    gemm_wmma_f32<<<dim3(MFH / 64, MB / 32), 32, 0, stream>>>(
        v0, MFH, fc1_w, MFH, fc1_b, v1, MFH, MFH, 1);
    gemm_wmma_f32<<<dim3(MFH / 64, MB / 32), 32, 0, stream>>>(
        v1, MFH, fc2_w, MFH, fc2_b, v0, MFH, MFH, 1);
    gemm_wmma_f32<<<dim3(MK / 64, MB / 32), 32, 0, stream>>>(
        v0, MFH, fc3_w, MFH, fc3_b, out, MK, MFH, 0);
}